// FBAMultiHeadAttention_6073083757090
// MI455X (gfx1250) — compile-verified
//
#include <hip/hip_runtime.h>
#include <math.h>

namespace {

constexpr int kE = 64, kH = 4, kD = 16, kNB = 8, kBS = 8, kS = 200;
constexpr int kRows = kBS * kS;        // 1600
constexpr float kScale = 0.25f;        // 1/sqrt(D)
constexpr float kEps = 1e-12f;

typedef __attribute__((ext_vector_type(2))) float v2f;
typedef __attribute__((ext_vector_type(8))) float v8f;

struct Ptrs {
  const float* input_m; const float* input_c; const float* mask;
  const int*   type_seq;
  const float* type_m;  const float* type_c;
  const float* rel_m;   const float* rel_c;
  const float* pos_m;   const float* pos_c;
  const float* P[38];   // params in dict order
  float *q1,*k1,*v1,*q2,*k2,*v2, *PQ,*PK, *RQ,*RK, *tab, *ctxm,*ctxc;
  float *out_hm,*out_hc,*out_probs;
};

// ---- fp32 WMMA 16x16x4 helpers ------------------------------------------
__device__ __forceinline__ v8f wmma4(v2f a, v2f b, v8f c) {
  return __builtin_amdgcn_wmma_f32_16x16x4_f32(false, a, false, b, (short)0, c,
                                               false, false);
}
// A tile (16x4) from row-major [.,ld]: lane<16 -> K={0,1}, lane>=16 -> K={2,3}
__device__ __forceinline__ v2f ldA(const float* p, int ld, int m0, int k0, int lane) {
  const float* q = p + (m0 + (lane & 15)) * ld + (k0 + ((lane >> 4) << 1));
  v2f r; r.x = q[0]; r.y = q[1]; return r;
}
// B tile (4x16) with B[k][n] = W[n][k]  (for x @ W^T), W row-major [.,ld]
__device__ __forceinline__ v2f ldBT(const float* W, int ld, int n0, int k0, int lane) {
  const float* q = W + (n0 + (lane & 15)) * ld + (k0 + ((lane >> 4) << 1));
  v2f r; r.x = q[0]; r.y = q[1]; return r;
}
// B tile (4x16) row-major: B[k][n] = M[(k0+k)*ld + n0+n]
__device__ __forceinline__ v2f ldB(const float* M, int ld, int k0, int n0, int lane) {
  const float* q = M + (k0 + ((lane >> 4) << 1)) * ld + (n0 + (lane & 15));
  v2f r; r.x = q[0]; r.y = q[ld]; return r;
}
// D tile store: VGPR i: lane<16 row m0+i, lane>=16 row m0+8+i
__device__ __forceinline__ void stD(float* out, int ld, int m0, int n0, int lane,
                                    v8f acc, int mlim) {
  int n = n0 + (lane & 15);
  int mb = m0 + ((lane >> 4) << 3);
#pragma unroll
  for (int i = 0; i < 8; ++i) {
    int m = mb + i;
    if (m < mlim) out[m * ld + n] = acc[i];
  }
}

// ---- Kernel 1: fused QKV projections  q1=xm_q(x_m)+bm_q(t_m), etc. ------
__device__ __forceinline__ void proj_one(const float* x, const float* t,
                                         const float* W1, const float* b1,
                                         const float* W2, const float* b2,
                                         float* out, bool elu,
                                         int r0, int n0, int lane) {
  v8f acc = {0.f,0.f,0.f,0.f,0.f,0.f,0.f,0.f};
#pragma unroll
  for (int k0 = 0; k0 < kE; k0 += 4) {
    acc = wmma4(ldA(x, kE, r0, k0, lane), ldBT(W1, kE, n0, k0, lane), acc);
    acc = wmma4(ldA(t, kE, r0, k0, lane), ldBT(W2, kE, n0, k0, lane), acc);
  }
  int n = n0 + (lane & 15);
  float bias = b1[n] + b2[n];
#pragma unroll
  for (int i = 0; i < 8; ++i) {
    float v = acc[i] + bias;
    if (elu) v = (v > 0.f) ? (v + 1.f) : __expf(v);   // elu(v)+1
    acc[i] = v;
  }
  stD(out, kE, r0, n0, lane, acc, kRows);
}

__global__ void proj_kernel(Ptrs P) {
  int lane = threadIdx.x & 31;
  int n0 = (threadIdx.x >> 5) * 16;   // 4 waves = 4 N tiles
  int r0 = blockIdx.x * 16;
  proj_one(P.input_m, P.type_m, P.P[0],  P.P[1],  P.P[12], P.P[13], P.q1, false, r0, n0, lane);
  proj_one(P.input_m, P.type_m, P.P[2],  P.P[3],  P.P[14], P.P[15], P.k1, false, r0, n0, lane);
  proj_one(P.input_m, P.type_m, P.P[4],  P.P[5],  P.P[16], P.P[17], P.v1, false, r0, n0, lane);
  proj_one(P.input_c, P.type_c, P.P[6],  P.P[7],  P.P[18], P.P[19], P.q2, true,  r0, n0, lane);
  proj_one(P.input_c, P.type_c, P.P[8],  P.P[9],  P.P[20], P.P[21], P.k2, true,  r0, n0, lane);
  proj_one(P.input_c, P.type_c, P.P[10], P.P[11], P.P[22], P.P[23], P.v2, true,  r0, n0, lane);
}

// ---- Kernel 2: per-head DxD linear on R / P tables ----------------------
// out[row, h*16+n] = sum_k src[row, h*16+k] * W[n][k] + b[n]
__global__ void headlin_kernel(const float* src, const float* W, const float* bias,
                               float* out, int nrows) {
  int lane = threadIdx.x & 31;
  int h = threadIdx.x >> 5;           // 4 waves = 4 heads
  int r0 = blockIdx.x * 16;
  v8f acc = {0.f,0.f,0.f,0.f,0.f,0.f,0.f,0.f};
#pragma unroll
  for (int k0 = 0; k0 < kD; k0 += 4)
    acc = wmma4(ldA(src, kE, r0, h * kD + k0, lane), ldBT(W, kD, 0, k0, lane), acc);
  float bb = bias[lane & 15];
#pragma unroll
  for (int i = 0; i < 8; ++i) acc[i] += bb;
  stD(out, kE, r0, h * kD, lane, acc, nrows);
}

// ---- Kernel 3: score table tab[b,h,i,t] = dist + cov_d ------------------
// scores[b,h,i,j] = -tab[b,h,i,ts[j]]*scale + mask  (j enters only via ts[j])
__global__ void tab_kernel(Ptrs P) {
  int tid = blockIdx.x * blockDim.x + threadIdx.x;
  if (tid >= kBS * kH * kS) return;
  int i = tid % kS;
  int h = (tid / kS) % kH;
  int b = tid / (kS * kH);
  int tsi = P.type_seq[b * kS + i];
  int ro = (b * kS + i) * kE + h * kD;                 // per-i row, this head
  int rb = (b * kNB + tsi) * kNB * kE + h * kD;        // rel table (t=0)
  float dist[kNB], covd[kNB];
#pragma unroll
  for (int t = 0; t < kNB; ++t) { dist[t] = 0.f; covd[t] = 0.f; }
  for (int d = 0; d < kD; ++d) {
    float m1q = P.q1[ro + d], m1k = P.k1[ro + d];
    float i1q = 1.f / fmaxf(P.q2[ro + d], 1e-24f);
    float i1k = 1.f / fmaxf(P.k2[ro + d], 1e-24f);
    float m3q = P.PQ[ro + d], m3k = P.PK[ro + d];
    float i3  = 1.f / fmaxf(P.pos_c[ro + d], 1e-24f);
    float aq = m1q * i1q + m3q * i3;
    float ak = m1k * i1k + m3k * i3;
    float sq = i1q + i3, sk = i1k + i3;
#pragma unroll
    for (int t = 0; t < kNB; ++t) {
      int off = rb + t * kE + d;
      float m2q = P.RQ[off], m2k = P.RK[off];
      float i2 = 1.f / fmaxf(P.rel_c[off], 1e-24f);
      float cq = 1.f / (sq + i2), ck = 1.f / (sk + i2);
      float mq = cq * (aq + m2q * i2);
      float mk = ck * (ak + m2k * i2);
      float dm = mq - mk;
      dist[t] += dm * dm;
      covd[t] += cq + ck - 2.f * sqrtf(cq) * sqrtf(ck);
    }
  }
  float* tr = P.tab + tid * kNB;
#pragma unroll
  for (int t = 0; t < kNB; ++t) tr[t] = dist[t] + covd[t];
}

// ---- Kernel 4: softmax over j, one wave32 per (b,h,i) row ---------------
__global__ void softmax_kernel(Ptrs P) {
  int lane = threadIdx.x & 31;
  int r = blockIdx.x * 8 + (threadIdx.x >> 5);  // (b*H+h)*S + i
  int i = r % kS;
  int b = r / (kS * kH);
  const float* tabr = P.tab + r * kNB;
  float tv[kNB];
#pragma unroll
  for (int t = 0; t < kNB; ++t) tv[t] = tabr[t];
  const int* tsb = P.type_seq + b * kS;
  const float* mrow = P.mask + (b * kS + i) * kS;  // mask[b,0,i,:]
  float sc[7];
  float mx = -3.4e38f;
  int cnt = 0;
  for (int j = lane; j < kS; j += 32) {
    float s = -tv[tsb[j]] * kScale + mrow[j];
    sc[cnt++] = s;
    mx = fmaxf(mx, s);
  }
#pragma unroll
  for (int off = 16; off > 0; off >>= 1) mx = fmaxf(mx, __shfl_xor(mx, off, 32));
  float sum = 0.f;
  cnt = 0;
  for (int j = lane; j < kS; j += 32) {
    float e = __expf(sc[cnt] - mx);
    sc[cnt] = e; sum += e; ++cnt;
  }
#pragma unroll
  for (int off = 16; off > 0; off >>= 1) sum += __shfl_xor(sum, off, 32);
  float inv = 1.f / sum;
  float* pr = P.out_probs + (size_t)r * kS;
  cnt = 0;
  for (int j = lane; j < kS; j += 32) pr[j] = sc[cnt++] * inv;
}

// ---- Kernel 5: ctx = probs @ v  (per b,h: [200x200]@[200x16]) -----------
__global__ void ctx_kernel(Ptrs P) {
  int lane = threadIdx.x & 31;
  int w = threadIdx.x >> 5;          // wave0: v1->ctx_m, wave1: v2->ctx_c
  int mt = blockIdx.x % 13;
  int bh = blockIdx.x / 13;
  int b = bh / kH, h = bh % kH;
  const float* probs = P.out_probs + (size_t)bh * kS * kS;
  const float* V = ((w == 0) ? P.v1 : P.v2) + b * kS * kE;
  float* out = ((w == 0) ? P.ctxm : P.ctxc) + b * kS * kE;
  int m0 = mt * 16;
  int m = m0 + (lane & 15); if (m >= kS) m = kS - 1;  // clamp reads, mask stores
  v8f acc = {0.f,0.f,0.f,0.f,0.f,0.f,0.f,0.f};
  for (int k0 = 0; k0 < kS; k0 += 4) {
    const float* pa = probs + m * kS + k0 + ((lane >> 4) << 1);
    v2f a; a.x = pa[0]; a.y = pa[1];
    acc = wmma4(a, ldB(V, kE, k0, h * kD, lane), acc);
  }
  stD(out, kE, m0, h * kD, lane, acc, kS);
}

// ---- Kernel 6: output projection + residual + layernorm -----------------
__global__ void out_kernel(Ptrs P) {
  __shared__ float lds[2][16][kE];
  int lane = threadIdx.x & 31;
  int w = threadIdx.x >> 5;          // wave0: hm, wave1: hc
  int r0 = blockIdx.x * 16;
  const float* ctx   = (w == 0) ? P.ctxm   : P.ctxc;
  const float* Wt    = (w == 0) ? P.P[32]  : P.P[34];   // meanW / covW
  const float* bt    = (w == 0) ? P.P[33]  : P.P[35];
  const float* resid = (w == 0) ? P.input_m : P.input_c;
  float* out         = (w == 0) ? P.out_hm : P.out_hc;
#pragma unroll
  for (int nt = 0; nt < 4; ++nt) {
    int n0 = nt * 16;
    v8f acc = {0.f,0.f,0.f,0.f,0.f,0.f,0.f,0.f};
#pragma unroll
    for (int k0 = 0; k0 < kE; k0 += 4)
      acc = wmma4(ldA(ctx, kE, r0, k0, lane), ldBT(Wt, kE, n0, k0, lane), acc);
    int n = n0 + (lane & 15);
    int mb = (lane >> 4) << 3;
#pragma unroll
    for (int i = 0; i < 8; ++i) {
      int m = mb + i;
      lds[w][m][n] = acc[i] + bt[n] + resid[(r0 + m) * kE + n];
    }
  }
  __syncthreads();
  if (lane < 16) {
    const float* row = lds[w][lane];
    float mu = 0.f;
    for (int c = 0; c < kE; ++c) mu += row[c];
    mu *= (1.f / kE);
    float var = 0.f;
    for (int c = 0; c < kE; ++c) { float d = row[c] - mu; var += d * d; }
    var *= (1.f / kE);
    float rstd = rsqrtf(var + kEps);
    const float* g = P.P[36]; const float* be = P.P[37];
    float* orow = out + (r0 + lane) * kE;
    for (int c = 0; c < kE; ++c) orow[c] = (row[c] - mu) * rstd * g[c] + be[c];
  }
}

} // namespace

extern "C" void kernel_launch(void* const* d_in, const int* in_sizes, int n_in,
                              void* d_out, int out_size, void* d_ws, size_t ws_size,
                              hipStream_t stream) {
  (void)in_sizes; (void)n_in; (void)out_size; (void)ws_size;
  Ptrs P;
  P.input_m  = (const float*)d_in[0];
  P.input_c  = (const float*)d_in[1];
  P.mask     = (const float*)d_in[2];
  P.type_seq = (const int*)  d_in[3];
  P.type_m   = (const float*)d_in[4];
  P.type_c   = (const float*)d_in[5];
  P.rel_m    = (const float*)d_in[6];
  P.rel_c    = (const float*)d_in[7];
  P.pos_m    = (const float*)d_in[8];
  P.pos_c    = (const float*)d_in[9];
  for (int k = 0; k < 38; ++k) P.P[k] = (const float*)d_in[10 + k];

  float* ws = (float*)d_ws;
  size_t o = 0;
  auto take = [&](size_t n) { float* p = ws + o; o += n; return p; };
  P.q1 = take(kRows * kE); P.k1 = take(kRows * kE); P.v1 = take(kRows * kE);
  P.q2 = take(kRows * kE); P.k2 = take(kRows * kE); P.v2 = take(kRows * kE);
  P.PQ = take(kRows * kE); P.PK = take(kRows * kE);
  P.RQ = take(kBS * kNB * kNB * kE); P.RK = take(kBS * kNB * kNB * kE);
  P.tab = take(kBS * kH * kS * kNB);
  P.ctxm = take(kRows * kE); P.ctxc = take(kRows * kE);

  P.out_hm    = (float*)d_out;
  P.out_hc    = P.out_hm + kRows * kE;
  P.out_probs = P.out_hc + kRows * kE;

  // 1) fused QKV projections (12 GEMMs, [1600x64]@[64x64]^T)
  proj_kernel<<<kRows / 16, 128, 0, stream>>>(P);
  // 2) R-table linears (Wq1/Wk1 on rel_m, rows = 8*64) and P linears (rows = 1600)
  headlin_kernel<<<(kBS * kNB * kNB) / 16, 128, 0, stream>>>(P.rel_m, P.P[24], P.P[25], P.RQ, kBS * kNB * kNB);
  headlin_kernel<<<(kBS * kNB * kNB) / 16, 128, 0, stream>>>(P.rel_m, P.P[28], P.P[29], P.RK, kBS * kNB * kNB);
  headlin_kernel<<<kRows / 16, 128, 0, stream>>>(P.pos_m, P.P[26], P.P[27], P.PQ, kRows);
  headlin_kernel<<<kRows / 16, 128, 0, stream>>>(P.pos_m, P.P[30], P.P[31], P.PK, kRows);
  // 3) score table [b,H,S,NB]
  tab_kernel<<<(kBS * kH * kS + 255) / 256, 256, 0, stream>>>(P);
  // 4) softmax + probs output
  softmax_kernel<<<(kBS * kH * kS) / 8, 256, 0, stream>>>(P);
  // 5) context GEMMs
  ctx_kernel<<<kBS * kH * 13, 64, 0, stream>>>(P);
  // 6) output projection + residual + layernorm
  out_kernel<<<kRows / 16, 64, 0, stream>>>(P);
}